// Net_24438363914899
// MI455X (gfx1250) — compile-verified
//
#include <hip/hip_runtime.h>

// CDNA5 / gfx1250 fused heterogeneous-GAT kernel.
// One workgroup (256 threads = 8 wave32) per batch element.
// Matmul H[t] = h[b] @ W[t] runs on V_WMMA_F32_16X16X4_F32 (full fp32).
// W is pre-transposed into d_ws so B fragments are single global_load_b64.

typedef __attribute__((ext_vector_type(2))) float v2f;
typedef __attribute__((ext_vector_type(8))) float v8f;

#define T_TYPES 3
#define IN_DIM  128
#define HID     128
#define N_NODES 32
#define N_ALLY  15
#define N_OPP   16
#define HP      132   // padded row length for Hs (132 % 64 == 4 -> conflict-free strided reads)

// One-time (per launch) transpose: WT[t][col][k] = W[t][k][col].  192 KB total.
__global__ void transpose_W_kernel(const float* __restrict__ W, float* __restrict__ WT)
{
    const int idx = blockIdx.x * 256 + threadIdx.x;      // < 3*128*128
    const int t   = idx >> 14;
    const int rem = idx & 16383;
    const int k   = rem >> 7;
    const int col = rem & 127;
    WT[(t << 14) + (col << 7) + k] = W[idx];
}

template <bool TRANSPOSED_W>
__launch_bounds__(256)
__global__ void het_gat_kernel(const float* __restrict__ h,
                               const unsigned char* __restrict__ mask, // jax bool -> 1 byte
                               const float* __restrict__ W,            // [t][k][col] or [t][col][k]
                               const float* __restrict__ a,
                               float* __restrict__ out,
                               int nB)
{
    __shared__ float Hs[T_TYPES][N_NODES][HP];   // ~50.7 KB: H[t][node][hid]
    __shared__ float aTs[T_TYPES][HID];          // a_top
    __shared__ float aBs[T_TYPES][HID];          // a_bot
    __shared__ float sS[9];                      // s_self[i][j]
    __shared__ float sA[T_TYPES * N_ALLY];       // s_ally[j][n]
    __shared__ float sO[T_TYPES * N_OPP];        // s_opp[j][n]
    __shared__ float wA[T_TYPES * N_ALLY];       // softmax weights (summed over i)
    __shared__ float wO[T_TYPES * N_OPP];

    const int tid  = threadIdx.x;
    const int b    = blockIdx.x;
    const size_t mstride = (size_t)nB * N_NODES;            // mask stride per type
    const size_t mbase   = (size_t)b * N_NODES;             // mask offset for this b

    // ---- stage attention vectors a_top / a_bot into LDS ----
    for (int r = tid; r < T_TYPES * 2 * HID; r += 256) {
        int t = r >> 8, hh = r & 255;
        float v = a[t * 2 * HID + hh];
        if (hh < HID) aTs[t][hh] = v;
        else          aBs[t][hh - HID] = v;
    }

    // ---- WMMA phase: Hs[t] = h[b](32x128) @ W[t](128x128) ----
    const int wave = tid >> 5;
    const int lane = tid & 31;
    const int half = lane >> 4;      // 0: lanes 0-15, 1: lanes 16-31
    const int lrow = lane & 15;
    const int koff = half * 2;       // A/B fragment K sub-offset per ISA 16x4 f32 layout
    const int col  = wave * 16 + lrow;   // each wave owns one 16-col slice of HID
    const float* hb = h + (size_t)b * N_NODES * IN_DIM;

    for (int t = 0; t < T_TYPES; ++t) {
        const float* Wt = W + (size_t)t * IN_DIM * HID;
        v8f c0 = {};   // nodes 0..15
        v8f c1 = {};   // nodes 16..31
        #pragma unroll 4
        for (int kk = 0; kk < 32; ++kk) {
            const int kA = kk * 4 + koff;
            // A fragments: lane holds h[b][row][kA], h[b][row][kA+1]
            v2f a0 = *(const v2f*)(hb + (size_t)lrow * IN_DIM + kA);
            v2f a1 = *(const v2f*)(hb + (size_t)(16 + lrow) * IN_DIM + kA);
            // B fragment: lane holds W[t][kA][col], W[t][kA+1][col]
            v2f bb;
            if (TRANSPOSED_W) {
                bb = *(const v2f*)(Wt + (size_t)col * IN_DIM + kA);   // one b64 load
            } else {
                bb.x = Wt[(size_t)kA * HID + col];
                bb.y = Wt[(size_t)(kA + 1) * HID + col];
            }
            c0 = __builtin_amdgcn_wmma_f32_16x16x4_f32(false, a0, false, bb,
                                                       (short)0, c0, false, false);
            c1 = __builtin_amdgcn_wmma_f32_16x16x4_f32(false, a1, false, bb,
                                                       (short)0, c1, false, false);
        }
        // C/D layout: VGPR r -> row r (lanes 0-15) / row r+8 (lanes 16-31)
        #pragma unroll
        for (int r = 0; r < 8; ++r) {
            const int mrow = half ? (r + 8) : r;
            Hs[t][mrow][col]      = c0[r];
            Hs[t][16 + mrow][col] = c1[r];
        }
    }
    __syncthreads();

    // ---- attention logits: 102 length-128 dot products ----
    if (tid < 102) {
        const float* rowp; const float* av; float* dst;
        if (tid < 9) {                       // s_self[i][j] = H[i][0][:] . a_top[j]
            int i = tid / 3, j = tid % 3;
            rowp = Hs[i][0]; av = aTs[j]; dst = &sS[tid];
        } else if (tid < 54) {               // s_ally[j][n] = H[j][1+n][:] . a_bot[j]
            int d = tid - 9, j = d / N_ALLY, n = d % N_ALLY;
            rowp = Hs[j][1 + n]; av = aBs[j]; dst = &sA[d];
        } else {                             // s_opp[j][n] = H[j][16+n][:] . a_bot[j]
            int d = tid - 54, j = d / N_OPP, n = d % N_OPP;
            rowp = Hs[j][16 + n]; av = aBs[j]; dst = &sO[d];
        }
        float s = 0.f;
        #pragma unroll 8
        for (int k = 0; k < HID; ++k) s += rowp[k] * av[k];
        *dst = s;
    }
    __syncthreads();

    // ---- masked softmaxes: wave 0 -> ally (135), wave 1 -> opp (144) ----
    if (wave == 0) {
        float mx = -__builtin_inff();
        for (int idx = lane; idx < 9 * N_ALLY; idx += 32) {
            int i = idx / (3 * N_ALLY), r2 = idx % (3 * N_ALLY);
            int j = r2 / N_ALLY, n = r2 % N_ALLY;
            if (mask[(size_t)j * mstride + mbase + 1 + n] == 0)
                mx = fmaxf(mx, sS[i * 3 + j] + sA[j * N_ALLY + n]);
        }
        for (int o = 16; o; o >>= 1) mx = fmaxf(mx, __shfl_xor(mx, o, 32));
        float z = 0.f;
        for (int idx = lane; idx < 9 * N_ALLY; idx += 32) {
            int i = idx / (3 * N_ALLY), r2 = idx % (3 * N_ALLY);
            int j = r2 / N_ALLY, n = r2 % N_ALLY;
            if (mask[(size_t)j * mstride + mbase + 1 + n] == 0)
                z += expf(sS[i * 3 + j] + sA[j * N_ALLY + n] - mx);
        }
        for (int o = 16; o; o >>= 1) z += __shfl_xor(z, o, 32);
        const float inv = 1.f / z;
        for (int p = lane; p < T_TYPES * N_ALLY; p += 32) {
            int j = p / N_ALLY, n = p % N_ALLY;
            float w = 0.f;
            if (mask[(size_t)j * mstride + mbase + 1 + n] == 0) {
                const float base = sA[p] - mx;
                for (int i = 0; i < 3; ++i) w += expf(sS[i * 3 + j] + base);
            }
            wA[p] = w * inv;
        }
    } else if (wave == 1) {
        float mx = -__builtin_inff();
        for (int idx = lane; idx < 9 * N_OPP; idx += 32) {
            int i = idx / (3 * N_OPP), r2 = idx % (3 * N_OPP);
            int j = r2 / N_OPP, n = r2 % N_OPP;
            if (mask[(size_t)j * mstride + mbase + 16 + n] == 0)
                mx = fmaxf(mx, sS[i * 3 + j] + sO[j * N_OPP + n]);
        }
        for (int o = 16; o; o >>= 1) mx = fmaxf(mx, __shfl_xor(mx, o, 32));
        float z = 0.f;
        for (int idx = lane; idx < 9 * N_OPP; idx += 32) {
            int i = idx / (3 * N_OPP), r2 = idx % (3 * N_OPP);
            int j = r2 / N_OPP, n = r2 % N_OPP;
            if (mask[(size_t)j * mstride + mbase + 16 + n] == 0)
                z += expf(sS[i * 3 + j] + sO[j * N_OPP + n] - mx);
        }
        for (int o = 16; o; o >>= 1) z += __shfl_xor(z, o, 32);
        const float inv = 1.f / z;
        for (int p = lane; p < T_TYPES * N_OPP; p += 32) {
            int j = p / N_OPP, n = p % N_OPP;
            float w = 0.f;
            if (mask[(size_t)j * mstride + mbase + 16 + n] == 0) {
                const float base = sO[p] - mx;
                for (int i = 0; i < 3; ++i) w += expf(sS[i * 3 + j] + base);
            }
            wO[p] = w * inv;
        }
    }
    __syncthreads();

    // ---- final weighted sum + ELU ----
    if (tid < HID) {
        float acc = 0.f;
        #pragma unroll
        for (int t = 0; t < T_TYPES; ++t)
            if (mask[(size_t)t * mstride + mbase] != 0)   // mask_self: True -> include
                acc += Hs[t][0][tid];
        for (int j = 0; j < T_TYPES; ++j) {
            #pragma unroll 5
            for (int n = 0; n < N_ALLY; ++n) acc += wA[j * N_ALLY + n] * Hs[j][1 + n][tid];
            #pragma unroll 4
            for (int n = 0; n < N_OPP; ++n)  acc += wO[j * N_OPP + n] * Hs[j][16 + n][tid];
        }
        out[(size_t)b * HID + tid] = (acc > 0.f) ? acc : expm1f(acc);
    }
}

extern "C" void kernel_launch(void* const* d_in, const int* in_sizes, int n_in,
                              void* d_out, int out_size, void* d_ws, size_t ws_size,
                              hipStream_t stream)
{
    (void)n_in; (void)out_size;
    const float*         h    = (const float*)d_in[0];
    const unsigned char* mask = (const unsigned char*)d_in[1]; // bool array
    // d_in[2]=num_ally, d_in[3]=num_opp, d_in[4]=self_type : compile-time constants here
    const float*         W    = (const float*)d_in[5];
    const float*         a    = (const float*)d_in[6];
    float*               out  = (float*)d_out;

    const int nB = in_sizes[0] / (N_NODES * IN_DIM);   // 8192
    const size_t wt_bytes = (size_t)T_TYPES * IN_DIM * HID * sizeof(float);  // 192 KB

    if (ws_size >= wt_bytes) {
        float* WT = (float*)d_ws;
        transpose_W_kernel<<<(T_TYPES * IN_DIM * HID) / 256, 256, 0, stream>>>(W, WT);
        het_gat_kernel<true><<<nB, 256, 0, stream>>>(h, mask, WT, a, out, nB);
    } else {
        het_gat_kernel<false><<<nB, 256, 0, stream>>>(h, mask, W, a, out, nB);
    }
}